// GCN_62388694941785
// MI455X (gfx1250) — compile-verified
//
#include <hip/hip_runtime.h>

#define B_  4
#define N_  20000
#define E_  320000
#define F_  128
#define M_  (B_ * N_)      // 80000 rows, divisible by 16
#define MF_ (M_ * F_)      // 10,240,000 elements

typedef float v2f __attribute__((ext_vector_type(2)));
typedef float v8f __attribute__((ext_vector_type(8)));
typedef unsigned int u32x4 __attribute__((ext_vector_type(4)));
typedef int          i32x4 __attribute__((ext_vector_type(4)));
typedef int          i32x8 __attribute__((ext_vector_type(8)));

// ---------------- degree / norm precompute ----------------

__global__ void k_deg_init(float* __restrict__ deg) {
    int i = blockIdx.x * blockDim.x + threadIdx.x;
    if (i < N_) deg[i] = 1.0f;  // self-loop contributes 1
}

__global__ void k_deg_count(const int* __restrict__ ei, float* __restrict__ deg) {
    int e = blockIdx.x * blockDim.x + threadIdx.x;
    if (e < E_) unsafeAtomicAdd(&deg[ei[E_ + e]], 1.0f);  // tgt = ei[1][e]
}

__global__ void k_dinv(float* __restrict__ deg) {
    int i = blockIdx.x * blockDim.x + threadIdx.x;
    if (i < N_) deg[i] = rsqrtf(deg[i]);  // deg >= 1 always
}

__global__ void k_norm(const int* __restrict__ ei, const float* __restrict__ dinv,
                       float* __restrict__ norm) {
    int e = blockIdx.x * blockDim.x + threadIdx.x;
    if (e < E_) norm[e] = dinv[ei[e]] * dinv[ei[E_ + e]];
}

// ---------------- TDM: DMA one 16x128 f32 tile into LDS with 132-dword stride ----
// D# per CDNA5 ISA ch.8: 2-D tile, data_size=4B, pad 4 DWORDs after every
// 128 DWORDs (pad_interval code 6 = 128 dw, pad_amount code 3 = 4 dw) so the
// LDS image lands at row stride 132 floats (bank-conflict-free for the WMMA
// fragment reads: 132 % 64 banks = 4-bank rotation per row).

__device__ __forceinline__ void tdm_load_tile(const float* gsrc, float* lds_dst) {
    const unsigned long long ga = (unsigned long long)(uintptr_t)gsrc;
    // Flat LDS pointers carry the LDS byte offset in addr[31:0] (aperture rule).
    const unsigned int lds = (unsigned int)(uintptr_t)lds_dst;

    u32x4 g0;
    g0.x = 1u;                                                 // count=1 (valid user D#)
    g0.y = lds;                                                // lds_addr [63:32]
    g0.z = (unsigned int)ga;                                   // global_addr[31:0]
    g0.w = (unsigned int)((ga >> 32) & 0x01FFFFFFu) | (2u << 30); // addr[56:32] | type=2

    i32x8 g1;
    g1[0] = (int)((2u << 16)   // data_size = 4 bytes
                | (1u << 20)   // pad_enable
                | (6u << 22)   // pad_interval: 128 DWORDs
                | (3u << 25)); // pad_amount: 4 DWORDs
    g1[1] = (int)(128u << 16); // tensor_dim0 = 128 (bits 79:48 low half)
    g1[2] = (int)(16u << 16);  // tensor_dim0 hi = 0 | tensor_dim1 = 16 (bits 111:80 low half)
    g1[3] = (int)(128u << 16); // tensor_dim1 hi = 0 | tile_dim0 = 128 (bits 127:112)
    g1[4] = 16;                // tile_dim1 = 16, tile_dim2 = 0
    g1[5] = 128;               // tensor_dim0_stride = 128 (bits 207:160 low)
    g1[6] = 0;                 // stride0 hi | tensor_dim1_stride low
    g1[7] = 0;                 // tensor_dim1_stride hi

    const i32x4 gz = {0, 0, 0, 0};
#if __clang_major__ >= 23
    const i32x8 gz8 = {0, 0, 0, 0, 0, 0, 0, 0};
    __builtin_amdgcn_tensor_load_to_lds(g0, g1, gz, gz, gz8, 0);
#else
    __builtin_amdgcn_tensor_load_to_lds(g0, g1, gz, gz, 0);
#endif
}

// ---------------- WMMA GEMM: H = X @ W, ACC = dinv^2 * H ----------------
// One block = one 16-row tile of X. 8 waves, each computes a 16x16 output tile
// (wave w -> columns [16w, 16w+16)). A tile DMA'd into LDS by the TDM; W
// streamed from global (64 KB, L2-resident). Fragment layouts per ISA 7.12.2
// (32-bit A 16x4 / B 4x16 / C 16x16).

__global__ __launch_bounds__(256) void k_gemm(const float* __restrict__ X,
                                              const float* __restrict__ W,
                                              const float* __restrict__ dinv,
                                              float* __restrict__ H,
                                              float* __restrict__ ACC) {
    __shared__ float As[16 * 132];
    const int row0 = blockIdx.x * 16;
    const int tid  = threadIdx.x;

    // Wave 0 issues the tile DMA (TDM ignores EXEC once issued; other waves
    // branch around it entirely). tensorcnt==0 for non-issuing waves.
    if (tid == 0) {
        tdm_load_tile(X + (size_t)row0 * F_, As);
    }
    __builtin_amdgcn_s_wait_tensorcnt(0);
    __syncthreads();

    const int lane = tid & 31;
    const int wave = tid >> 5;       // 0..7
    const int col0 = wave * 16;
    const int half = lane >> 4;      // lanes 16-31 hold K+2 / M+8 halves
    const int l15  = lane & 15;

    v8f c = {};
    #pragma unroll
    for (int k0 = 0; k0 < F_; k0 += 4) {
        const int ka = k0 + 2 * half;
        v2f a, b;
        // A 16x4: lane(l15) = M, v0 = K=ka, v1 = K=ka+1
        a.x = As[l15 * 132 + ka];
        a.y = As[l15 * 132 + ka + 1];
        // B 4x16: lane(l15) = N, v0 = K=ka row, v1 = K=ka+1 row
        b.x = W[ka * F_ + col0 + l15];
        b.y = W[(ka + 1) * F_ + col0 + l15];
        c = __builtin_amdgcn_wmma_f32_16x16x4_f32(false, a, false, b,
                                                  (short)0, c, false, false);
    }

    // C/D layout: VGPR r -> M = r (lanes 0-15) or r+8 (lanes 16-31), N = l15
    #pragma unroll
    for (int r = 0; r < 8; ++r) {
        const int row = row0 + r + 8 * half;
        const int col = col0 + l15;
        const float dv = dinv[row % N_];
        const float h  = c[r];
        H[row * F_ + col]   = h;
        ACC[row * F_ + col] = dv * dv * h;   // self-loop term initializes ACC
    }
}

// ---------------- edge scatter: ACC[b,tgt,:] += norm[e] * H[b,src,:] ----------------
// One wave per (edge, batch); lane owns 4 consecutive features.
// 512B coalesced float4 gather per wave + 4 fp32 global atomics per lane.
// H and ACC are both L2-resident (41 MB each < 192 MB), so the RMW traffic
// stays on-chip.

__global__ __launch_bounds__(256) void k_scatter(const int* __restrict__ ei,
                                                 const float* __restrict__ norm,
                                                 const float* __restrict__ H,
                                                 float* __restrict__ ACC) {
    const int gw = blockIdx.x * 8 + (threadIdx.x >> 5);   // global wave id
    if (gw >= E_ * B_) return;
    const int lane = threadIdx.x & 31;
    const int e = gw % E_;
    const int b = gw / E_;
    const int s = ei[e];
    const int t = ei[E_ + e];
    const float nm = norm[e];

    const float4* hp = (const float4*)&H[((size_t)b * N_ + s) * F_];
    const float4 v = hp[lane];
    float* ap = &ACC[((size_t)b * N_ + t) * F_ + lane * 4];
    unsafeAtomicAdd(ap + 0, nm * v.x);
    unsafeAtomicAdd(ap + 1, nm * v.y);
    unsafeAtomicAdd(ap + 2, nm * v.z);
    unsafeAtomicAdd(ap + 3, nm * v.w);
}

// ---------------- bias + ReLU ----------------

__global__ void k_bias_relu(const float* __restrict__ ACC,
                            const float* __restrict__ bias,
                            float* __restrict__ OUT) {
    int i = blockIdx.x * blockDim.x + threadIdx.x;   // grid sized exactly to MF_
    float v = ACC[i] + bias[i & 127];
    OUT[i] = v > 0.0f ? v : 0.0f;
}

// ---------------- launcher ----------------

extern "C" void kernel_launch(void* const* d_in, const int* in_sizes, int n_in,
                              void* d_out, int out_size, void* d_ws, size_t ws_size,
                              hipStream_t stream) {
    const float* x    = (const float*)d_in[0];
    const int*   ei   = (const int*)  d_in[1];
    const float* Wt[3] = {(const float*)d_in[2], (const float*)d_in[3], (const float*)d_in[4]};
    const float* bs[3] = {(const float*)d_in[5], (const float*)d_in[6], (const float*)d_in[7]};
    float* out = (float*)d_out;

    float* ws  = (float*)d_ws;
    float* Xb  = ws;                       // next-layer input     (MF_)
    float* Hb  = ws + (size_t)MF_;         // GEMM output H        (MF_)
    float* ACC = ws + 2 * (size_t)MF_;     // scatter accumulator  (MF_)
    float* DEG = ws + 3 * (size_t)MF_;     // deg -> dinv in place (N_)
    float* NRM = DEG + N_;                 // per-edge norm        (E_)

    // normalization precompute (recomputed every call: deterministic)
    k_deg_init <<<(N_ + 255) / 256, 256, 0, stream>>>(DEG);
    k_deg_count<<<(E_ + 255) / 256, 256, 0, stream>>>(ei, DEG);
    k_dinv     <<<(N_ + 255) / 256, 256, 0, stream>>>(DEG);
    k_norm     <<<(E_ + 255) / 256, 256, 0, stream>>>(ei, DEG, NRM);

    const float* cur = x;
    for (int l = 0; l < 3; ++l) {
        k_gemm   <<<M_ / 16,        256, 0, stream>>>(cur, Wt[l], DEG, Hb, ACC);
        k_scatter<<<(E_ * B_) / 8,  256, 0, stream>>>(ei, NRM, Hb, ACC);
        float* dst = (l == 2) ? out : Xb;
        k_bias_relu<<<MF_ / 256,    256, 0, stream>>>(ACC, bs[l], dst);
        cur = dst;
    }
}